// Loss_87256555586246
// MI455X (gfx1250) — compile-verified
//
#include <hip/hip_runtime.h>
#include <hip/hip_bf16.h>
#include <math.h>

// SSD MultiBox loss: B=32, N=8732, C=81
#define BN   279424      // B*N anchors
#define CCLS 81

typedef __attribute__((ext_vector_type(2))) float v2f;
typedef __attribute__((ext_vector_type(8))) float v8f;

// workspace layout (unsigned int / float words)
enum {
  WS_POS   = 0,   // uint  pos_num
  WS_LOSSL = 1,   // float loc loss sum
  WS_CPOS  = 2,   // float CE sum over positives
  WS_CNEG  = 3,   // float CE sum over selected negatives
  WS_T     = 4,   // uint  threshold high-16 bin (0xFFFFFFFF = none)
  WS_REMT  = 5,   // uint  count needed inside bin T
  WS_V     = 6,   // uint  exact 32-bit threshold bits
  WS_REM2  = 7,   // uint  count to take at bits == V
  WS_TIE   = 8,   // uint  tie ticket counter
  WS_HIST1 = 16,                  // 65536 bins (high 16 bits)
  WS_HIST2 = 16 + 65536,          // 65536 bins (low 16 bits of bin T)
  WS_ZERO_WORDS = 16 + 2 * 65536, // words zeroed by init = 131088
  WS_LOSSC = WS_ZERO_WORDS,       // BN floats: per-anchor CE loss
  WS_BITS  = WS_ZERO_WORDS + BN   // BN uints: con_neg bits
};

__device__ __forceinline__ float smooth_l1(float x) {
  float ax = fabsf(x);
  return (ax < 1.0f) ? 0.5f * ax * ax : ax - 0.5f;
}

__global__ void k_init(unsigned int* __restrict__ ws) {
  int i = blockIdx.x * 256 + threadIdx.x;
  if (i < WS_ZERO_WORDS) ws[i] = 0u;
}

// One wave32 per anchor: log-softmax CE, mask, histogram of con_neg bits.
__global__ void k_ce(const float* __restrict__ plabel,
                     const int*   __restrict__ glabel,
                     unsigned int* __restrict__ ws) {
  float* wsf = (float*)ws;
  float* loss_c      = (float*)(ws + WS_LOSSC);
  unsigned int* bits = ws + WS_BITS;
  unsigned int* h1   = ws + WS_HIST1;

  int lane   = threadIdx.x & 31;
  int wave   = threadIdx.x >> 5;
  int anchor = blockIdx.x * 8 + wave;              // grid sized exactly

  const float* row = plabel + (size_t)anchor * CCLS;
  float x0 = row[lane];
  float x1 = row[32 + lane];
  float x2 = (lane < CCLS - 64) ? row[64 + lane] : -INFINITY;

  float m = fmaxf(fmaxf(x0, x1), x2);
  #pragma unroll
  for (int o = 16; o > 0; o >>= 1) m = fmaxf(m, __shfl_xor(m, o, 32));

  float s = expf(x0 - m) + expf(x1 - m) + ((lane < CCLS - 64) ? expf(x2 - m) : 0.0f);
  #pragma unroll
  for (int o = 16; o > 0; o >>= 1) s += __shfl_xor(s, o, 32);

  __shared__ float s_closs[8];
  __shared__ int   s_pos[8];

  if (lane == 0) {
    int gt = glabel[anchor];
    float lc = (m + logf(s)) - row[gt];            // -log_softmax[gt] >= 0
    loss_c[anchor] = lc;
    bool pos = gt > 0;
    float con = pos ? 0.0f : fmaxf(lc, 0.0f);      // clamp fp noise
    unsigned int cb = __float_as_uint(con);        // nonneg float: bit order == value order
    bits[anchor] = cb;
    atomicAdd(&h1[cb >> 16], 1u);
    s_closs[wave] = pos ? lc : 0.0f;
    s_pos[wave]   = pos ? 1 : 0;
  }
  __syncthreads();
  if (threadIdx.x == 0) {
    float c = 0.0f; int p = 0;
    #pragma unroll
    for (int w = 0; w < 8; ++w) { c += s_closs[w]; p += s_pos[w]; }
    if (c != 0.0f) atomicAdd(&wsf[WS_CPOS], c);
    if (p)         atomicAdd(&ws[WS_POS], (unsigned int)p);
  }
}

// One wave32 per 16 anchors: masked SmoothL1 over the 4 box coords, row-summed
// via V_WMMA_F32_16X16X4_F32 (A = 16x4 masked smooth-L1 values, B = ones).
// Mask is applied to A *before* the WMMA so the extraction is branch-free and
// EXEC stays all-ones across the whole wave body.
__global__ void k_loc(const float* __restrict__ ploc,
                      const float* __restrict__ gloc,
                      const int*   __restrict__ glabel,
                      unsigned int* __restrict__ ws) {
  float* wsf = (float*)ws;
  int lane = threadIdx.x & 31;
  int wave = threadIdx.x >> 5;
  int tile = blockIdx.x * 8 + wave;                // grid sized exactly
  int base = tile * 16;
  int mrow = lane & 15;                            // M index (anchor within tile)
  int half = lane >> 4;                            // 0: K=0,1  1: K=2,3
  int a    = base + mrow;

  // per ISA A-layout for 16x4 f32: lane holds K = 2*half + {0,1}
  const float2* p2 = (const float2*)ploc;
  const float2* g2 = (const float2*)gloc;
  float2 p = p2[a * 2 + half];
  float2 g = g2[a * 2 + half];
  float msk = (glabel[a] > 0) ? 1.0f : 0.0f;       // coalesced, uniform per anchor

  v2f A; A.x = msk * smooth_l1(p.x - g.x); A.y = msk * smooth_l1(p.y - g.y);
  v2f Bv; Bv.x = 1.0f; Bv.y = 1.0f;                // ones: B layout-invariant
  v8f Cv = {};
  // D[i][j] = masked rowsum_i for every column j
  v8f D = __builtin_amdgcn_wmma_f32_16x16x4_f32(
      false, A, false, Bv, (short)0, Cv, false, false);

  // D layout: VGPR r -> (lanes 0-15: M=r, lanes 16-31: M=8+r); all columns equal.
  float part = 0.0f;
  #pragma unroll
  for (int j = 0; j < 8; ++j) part += D[j];        // rows 0-7 (lo lanes) / 8-15 (hi lanes)
  float tot = __shfl(part, 0, 32) + __shfl(part, 16, 32);  // tile total, uniform

  __shared__ float sh[8];
  if (lane == 0) sh[wave] = tot;
  __syncthreads();
  if (threadIdx.x == 0) {
    float t = 0.0f;
    #pragma unroll
    for (int w = 0; w < 8; ++w) t += sh[w];
    if (t != 0.0f) atomicAdd(&wsf[WS_LOSSL], t);
  }
}

// Single block: find high-16 threshold bin T and remainder needed in it.
__global__ void k_sel1(unsigned int* __restrict__ ws) {
  __shared__ unsigned int chunk[256];
  const unsigned int* h = ws + WS_HIST1;
  int t = threadIdx.x;
  unsigned int sum = 0;
  for (int j = 0; j < 256; ++j) sum += h[t * 256 + j];
  chunk[t] = sum;
  __syncthreads();
  if (t == 0) {
    unsigned long long k64 = 3ull * ws[WS_POS];
    unsigned int k = (k64 > (unsigned long long)BN) ? BN : (unsigned int)k64;
    unsigned int T = 0xFFFFFFFFu, rem = 0;
    if (k > 0) {
      unsigned int cum = 0;
      for (int c = 255; c >= 0; --c) {
        if (cum + chunk[c] >= k) {
          for (int b = c * 256 + 255; b >= c * 256; --b) {
            cum += h[b];
            if (cum >= k) { T = (unsigned int)b; rem = k - (cum - h[b]); break; }
          }
          break;
        }
        cum += chunk[c];
      }
    }
    ws[WS_T] = T; ws[WS_REMT] = rem;
  }
}

__global__ void k_hist2(unsigned int* __restrict__ ws) {
  const unsigned int* bits = ws + WS_BITS;
  unsigned int* h2 = ws + WS_HIST2;
  unsigned int T = ws[WS_T];
  int i = blockIdx.x * 256 + threadIdx.x;
  if (i < BN) {
    unsigned int b = bits[i];
    if ((b >> 16) == T) atomicAdd(&h2[b & 0xFFFFu], 1u);
  }
}

// Single block: exact 32-bit threshold V and tie remainder rem2.
__global__ void k_sel2(unsigned int* __restrict__ ws) {
  __shared__ unsigned int chunk[256];
  const unsigned int* h = ws + WS_HIST2;
  int t = threadIdx.x;
  unsigned int sum = 0;
  for (int j = 0; j < 256; ++j) sum += h[t * 256 + j];
  chunk[t] = sum;
  __syncthreads();
  if (t == 0) {
    unsigned int T = ws[WS_T], rem = ws[WS_REMT];
    unsigned int V = 0xFFFFFFFFu, rem2 = 0;
    if (T != 0xFFFFFFFFu && rem > 0) {
      unsigned int cum = 0;
      for (int c = 255; c >= 0; --c) {
        if (cum + chunk[c] >= rem) {
          for (int b = c * 256 + 255; b >= c * 256; --b) {
            cum += h[b];
            if (cum >= rem) {
              V = (T << 16) | (unsigned int)b;
              rem2 = rem - (cum - h[b]);
              break;
            }
          }
          break;
        }
        cum += chunk[c];
      }
    }
    ws[WS_V] = V; ws[WS_REM2] = rem2;
  }
}

// Sum CE loss over selected negatives (bits > V, plus rem2 tickets at == V).
__global__ void k_neg(unsigned int* __restrict__ ws) {
  float* wsf = (float*)ws;
  const float* loss_c        = (const float*)(ws + WS_LOSSC);
  const unsigned int* bits   = ws + WS_BITS;
  unsigned int V = ws[WS_V], rem2 = ws[WS_REM2];
  int i = blockIdx.x * 256 + threadIdx.x;
  float val = 0.0f;
  if (i < BN) {
    unsigned int b = bits[i];
    bool sel = (b > V) ||
               (b == V && rem2 > 0 && atomicAdd(&ws[WS_TIE], 1u) < rem2);
    if (sel) val = loss_c[i];   // equal-bit ties contribute equal loss: exact
  }
  #pragma unroll
  for (int o = 16; o > 0; o >>= 1) val += __shfl_xor(val, o, 32);
  __shared__ float sh[8];
  int lane = threadIdx.x & 31, wave = threadIdx.x >> 5;
  if (lane == 0) sh[wave] = val;
  __syncthreads();
  if (threadIdx.x == 0) {
    float t = 0.0f;
    #pragma unroll
    for (int w = 0; w < 8; ++w) t += sh[w];
    if (t != 0.0f) atomicAdd(&wsf[WS_CNEG], t);
  }
}

__global__ void k_fin(const unsigned int* __restrict__ ws, float* __restrict__ out) {
  const float* wsf = (const float*)ws;
  unsigned int pos = ws[WS_POS];
  float tot = wsf[WS_LOSSL] + wsf[WS_CPOS] + wsf[WS_CNEG];
  out[0] = (pos > 0) ? tot / (float)pos : 0.0f;
}

extern "C" void kernel_launch(void* const* d_in, const int* in_sizes, int n_in,
                              void* d_out, int out_size, void* d_ws, size_t ws_size,
                              hipStream_t stream) {
  const float* ploc   = (const float*)d_in[0];
  const float* plabel = (const float*)d_in[1];
  const float* gloc   = (const float*)d_in[2];
  const int*   glabel = (const int*)d_in[3];
  float* out = (float*)d_out;
  unsigned int* ws = (unsigned int*)d_ws;

  (void)in_sizes; (void)n_in; (void)out_size; (void)ws_size;

  k_init <<<(WS_ZERO_WORDS + 255) / 256, 256, 0, stream>>>(ws);
  k_ce   <<<BN / 8,        256, 0, stream>>>(plabel, glabel, ws);     // 1 wave/anchor
  k_loc  <<<BN / 16 / 8,   256, 0, stream>>>(ploc, gloc, glabel, ws); // 1 wave/16 anchors (WMMA)
  k_sel1 <<<1,             256, 0, stream>>>(ws);
  k_hist2<<<(BN + 255) / 256, 256, 0, stream>>>(ws);
  k_sel2 <<<1,             256, 0, stream>>>(ws);
  k_neg  <<<(BN + 255) / 256, 256, 0, stream>>>(ws);
  k_fin  <<<1, 1, 0, stream>>>(ws, out);
}